// LatentPyramid_39410619908165
// MI455X (gfx1250) — compile-verified
//
#include <hip/hip_runtime.h>

typedef float v4f __attribute__((ext_vector_type(4)));

#define Q_STEP 0.0625f           // 1/16
#define NPTS   2097152
#define NC0    12
#define RES0   1024
#define HW0    (RES0 * RES0)
#define NC1    20
#define RES1   512
#define HW1    (RES1 * RES1)
#define OUTC   68                // 4*NC0 + NC1

__device__ __forceinline__ int clampi(int x, int lo, int hi) {
    return x < lo ? lo : (x > hi ? hi : x);
}

// ---------------------------------------------------------------------------
// Pass 1 (workspace path): transpose-quantize [C,H,W] -> [H,W,C] AoS.
//   qT[p*C + c] = g[c*HW + p] + (noise[c*HW + p] - 0.5) * Q
// Raw g/noise are read once -> NT loads (stay out of L2). The AoS result is
// written with default RT policy so the 68 MB hot set stays L2-resident
// (192 MB global L2) for the gather pass.
// One thread per pixel: per-channel reads are lane-coalesced; the 48/80-byte
// AoS row is written with aligned b128 stores.
// ---------------------------------------------------------------------------
template <int C, int HW>
__global__ __launch_bounds__(256) void quantT_kernel(
    const float* __restrict__ g, const float* __restrict__ n,
    float* __restrict__ qT)
{
    int p = blockIdx.x * blockDim.x + threadIdx.x;
    if (p >= HW) return;
    float vals[C];
#pragma unroll
    for (int c = 0; c < C; ++c) {
        float gv = __builtin_nontemporal_load(g + (size_t)c * HW + p);
        float nv = __builtin_nontemporal_load(n + (size_t)c * HW + p);
        vals[c] = gv + (nv - 0.5f) * Q_STEP;
    }
    v4f* dst = (v4f*)(qT + (size_t)p * C);   // C*4 % 16 == 0 for C=12,20
#pragma unroll
    for (int k = 0; k < C / 4; ++k) {
        v4f w = {vals[4 * k], vals[4 * k + 1], vals[4 * k + 2], vals[4 * k + 3]};
        dst[k] = w;
    }
}

// ---------------------------------------------------------------------------
// Gather pass over transposed AoS planes (workspace path).
// feat0: 12 b128 loads sharing ONE address (corner deltas fit the 24-bit
// instruction offset: +48 B, +49152 B, +49200 B).
// feat1: 4 tap addresses x 5 b128 loads (imm 0..64 B).
// Output: 17 aligned non-temporal b128 stores (272 B/point, 16-B aligned).
// ---------------------------------------------------------------------------
__global__ __launch_bounds__(256) void gatherT_kernel(
    const float* __restrict__ uv,
    const float* __restrict__ q0t,   // [RES0, RES0, NC0]
    const float* __restrict__ q1t,   // [RES1, RES1, NC1]
    float* __restrict__ out)
{
    int pid = blockIdx.x * blockDim.x + threadIdx.x;
    if (pid >= NPTS) return;

    float u = __builtin_nontemporal_load(uv + 2 * (size_t)pid);
    float v = __builtin_nontemporal_load(uv + 2 * (size_t)pid + 1);

    float* orow = out + (size_t)pid * OUTC;
    auto stv = [&](int k, v4f w) {
        __builtin_nontemporal_store(w, ((v4f*)orow) + k);   // th:TH_STORE_NT
    };

    // ---- high-res 4-neighbor gather (clamped), 4*12 channels -------------
    {
        float px = u * (float)RES0 - 0.5f;
        float py = v * (float)RES0 - 0.5f;
        int ax0 = clampi((int)floorf(px), 0, RES0 - 2);
        int ay0 = clampi((int)floorf(py), 0, RES0 - 2);
        const v4f* b = (const v4f*)(q0t + (size_t)(ay0 * RES0 + ax0) * NC0);
        // corner (y0,x0): b[0..2]; (y0,x1): b[3..5]
        // corner (y1,x0): b[3*RES0 .. +2]; (y1,x1): b[3*RES0+3 .. +5]
        stv(0, b[0]);  stv(1, b[1]);  stv(2, b[2]);        // ch 0..11
        stv(3, b[3]);  stv(4, b[4]);  stv(5, b[5]);        // ch 12..23
        const v4f* br = b + 3 * RES0;
        stv(6, br[0]); stv(7, br[1]); stv(8, br[2]);       // ch 24..35
        stv(9, br[3]); stv(10, br[4]); stv(11, br[5]);     // ch 36..47
    }

    // ---- low-res bilinear, zero padding, 20 channels ---------------------
    {
        float qx = u * (float)RES1 - 0.5f;
        float qy = v * (float)RES1 - 0.5f;
        float fx = floorf(qx), fy = floorf(qy);
        float wx = qx - fx,    wy = qy - fy;
        int x0 = (int)fx, y0 = (int)fy;
        int x1 = x0 + 1,  y1 = y0 + 1;

        float vx0 = (x0 >= 0 && x0 < RES1) ? 1.f : 0.f;
        float vx1 = (x1 >= 0 && x1 < RES1) ? 1.f : 0.f;
        float vy0 = (y0 >= 0 && y0 < RES1) ? 1.f : 0.f;
        float vy1 = (y1 >= 0 && y1 < RES1) ? 1.f : 0.f;

        // taps are finite -> folding validity into weights == zeroing taps
        float w00 = (1.f - wy) * (1.f - wx) * vy0 * vx0;
        float w01 = (1.f - wy) * wx         * vy0 * vx1;
        float w10 = wy         * (1.f - wx) * vy1 * vx0;
        float w11 = wy         * wx         * vy1 * vx1;

        int cx0 = clampi(x0, 0, RES1 - 1), cx1 = clampi(x1, 0, RES1 - 1);
        int cy0 = clampi(y0, 0, RES1 - 1), cy1 = clampi(y1, 0, RES1 - 1);
        const v4f* t00 = (const v4f*)(q1t + (size_t)(cy0 * RES1 + cx0) * NC1);
        const v4f* t01 = (const v4f*)(q1t + (size_t)(cy0 * RES1 + cx1) * NC1);
        const v4f* t10 = (const v4f*)(q1t + (size_t)(cy1 * RES1 + cx0) * NC1);
        const v4f* t11 = (const v4f*)(q1t + (size_t)(cy1 * RES1 + cx1) * NC1);

#pragma unroll
        for (int k = 0; k < NC1 / 4; ++k) {
            v4f acc = t00[k] * w00 + t01[k] * w01 + t10[k] * w10 + t11[k] * w11;
            stv(12 + k, acc);
        }
    }
}

// ---------------------------------------------------------------------------
// Fallback (no workspace): fused per-tap noise, channel-major b32 gathers.
// ---------------------------------------------------------------------------
__global__ __launch_bounds__(256) void gather_fused_kernel(
    const float* __restrict__ uv,
    const float* __restrict__ p0, const float* __restrict__ n0,
    const float* __restrict__ p1, const float* __restrict__ n1,
    float* __restrict__ out)
{
    int pid = blockIdx.x * blockDim.x + threadIdx.x;
    if (pid >= NPTS) return;

    float u = __builtin_nontemporal_load(uv + 2 * (size_t)pid);
    float v = __builtin_nontemporal_load(uv + 2 * (size_t)pid + 1);

    float* orow = out + (size_t)pid * OUTC;
    auto st4 = [&](int k, float a, float b, float c, float d) {
        v4f w = {a, b, c, d};
        __builtin_nontemporal_store(w, ((v4f*)orow) + k);
    };
    auto sample = [&](const float* __restrict__ g,
                      const float* __restrict__ n, int idx) -> float {
        return g[idx] + (n[idx] - 0.5f) * Q_STEP;
    };

    float px = u * (float)RES0 - 0.5f;
    float py = v * (float)RES0 - 0.5f;
    int ax0 = clampi((int)floorf(px), 0, RES0 - 2);
    int ay0 = clampi((int)floorf(py), 0, RES0 - 2);
    int b00 = ay0 * RES0 + ax0;
    int b10 = b00 + RES0;

    {
        float t[NC0], s[NC0];
#pragma unroll
        for (int c = 0; c < NC0; ++c) {
            int o = c * HW0;
            t[c] = sample(p0, n0, o + b00);
            s[c] = sample(p0, n0, o + b00 + 1);
        }
        st4(0, t[0], t[1], t[2], t[3]);   st4(1, t[4], t[5], t[6], t[7]);
        st4(2, t[8], t[9], t[10], t[11]); st4(3, s[0], s[1], s[2], s[3]);
        st4(4, s[4], s[5], s[6], s[7]);   st4(5, s[8], s[9], s[10], s[11]);
#pragma unroll
        for (int c = 0; c < NC0; ++c) {
            int o = c * HW0;
            t[c] = sample(p0, n0, o + b10);
            s[c] = sample(p0, n0, o + b10 + 1);
        }
        st4(6, t[0], t[1], t[2], t[3]);   st4(7, t[4], t[5], t[6], t[7]);
        st4(8, t[8], t[9], t[10], t[11]); st4(9, s[0], s[1], s[2], s[3]);
        st4(10, s[4], s[5], s[6], s[7]);  st4(11, s[8], s[9], s[10], s[11]);
    }
    {
        float qx = u * (float)RES1 - 0.5f;
        float qy = v * (float)RES1 - 0.5f;
        float fx = floorf(qx), fy = floorf(qy);
        float wx = qx - fx,    wy = qy - fy;
        int x0 = (int)fx, y0 = (int)fy;
        int x1 = x0 + 1,  y1 = y0 + 1;
        float vx0 = (x0 >= 0 && x0 < RES1) ? 1.f : 0.f;
        float vx1 = (x1 >= 0 && x1 < RES1) ? 1.f : 0.f;
        float vy0 = (y0 >= 0 && y0 < RES1) ? 1.f : 0.f;
        float vy1 = (y1 >= 0 && y1 < RES1) ? 1.f : 0.f;
        float w00 = (1.f - wy) * (1.f - wx) * vy0 * vx0;
        float w01 = (1.f - wy) * wx         * vy0 * vx1;
        float w10 = wy         * (1.f - wx) * vy1 * vx0;
        float w11 = wy         * wx         * vy1 * vx1;
        int cx0 = clampi(x0, 0, RES1 - 1), cx1 = clampi(x1, 0, RES1 - 1);
        int cy0 = clampi(y0, 0, RES1 - 1), cy1 = clampi(y1, 0, RES1 - 1);
        int i00 = cy0 * RES1 + cx0, i01 = cy0 * RES1 + cx1;
        int i10 = cy1 * RES1 + cx0, i11 = cy1 * RES1 + cx1;

        float r[NC1];
#pragma unroll
        for (int c = 0; c < NC1; ++c) {
            int o = c * HW1;
            r[c] = sample(p1, n1, o + i00) * w00
                 + sample(p1, n1, o + i01) * w01
                 + sample(p1, n1, o + i10) * w10
                 + sample(p1, n1, o + i11) * w11;
        }
        st4(12, r[0],  r[1],  r[2],  r[3]);
        st4(13, r[4],  r[5],  r[6],  r[7]);
        st4(14, r[8],  r[9],  r[10], r[11]);
        st4(15, r[12], r[13], r[14], r[15]);
        st4(16, r[16], r[17], r[18], r[19]);
    }
}

extern "C" void kernel_launch(void* const* d_in, const int* in_sizes, int n_in,
                              void* d_out, int out_size, void* d_ws, size_t ws_size,
                              hipStream_t stream)
{
    const float* uv = (const float*)d_in[0];
    const float* g0 = (const float*)d_in[1];
    const float* g1 = (const float*)d_in[2];
    const float* n0 = (const float*)d_in[3];
    const float* n1 = (const float*)d_in[4];
    // d_in[5] = mip (unused by the reference math)
    float* out = (float*)d_out;

    const size_t q0_elems = (size_t)NC0 * HW0;   // 12,582,912
    const size_t q1_elems = (size_t)NC1 * HW1;   //  5,242,880
    const size_t need = (q0_elems + q1_elems) * sizeof(float);  // ~68 MB

    if (ws_size >= need) {
        float* q0t = (float*)d_ws;
        float* q1t = q0t + q0_elems;
        quantT_kernel<NC0, HW0><<<HW0 / 256, 256, 0, stream>>>(g0, n0, q0t);
        quantT_kernel<NC1, HW1><<<HW1 / 256, 256, 0, stream>>>(g1, n1, q1t);
        gatherT_kernel<<<NPTS / 256, 256, 0, stream>>>(uv, q0t, q1t, out);
    } else {
        gather_fused_kernel<<<NPTS / 256, 256, 0, stream>>>(
            uv, g0, n0, g1, n1, out);
    }
}